// Raycaster_22101901705954
// MI455X (gfx1250) — compile-verified
//
#include <hip/hip_runtime.h>
#include <math.h>

#define RW 320
#define RH 240
#define NSAMP 256
#define VRES 320
#define VRES2 (VRES * VRES)

typedef __attribute__((ext_vector_type(4))) float v4f;  // native vector: OK for nontemporal builtin

__device__ __forceinline__ float lerpf(float a, float b, float f) {
    return fmaf(f, b - a, a);
}

__device__ __forceinline__ float clampf(float v, float lo, float hi) {
    return fminf(fmaxf(v, lo), hi);   // -> v_med3_f32
}

__global__ __launch_bounds__(256) void Raycaster_22101901705954_kernel(
    const float* __restrict__ vol,     // [320,320,320]
    const float* __restrict__ tf,      // [128,4]
    const float* __restrict__ cam,     // [3]
    float* __restrict__ out)           // [320,240,4]
{
    // ---- cache the 2KB transfer function in LDS (ds_store_b128 / ds_load_b128)
    __shared__ v4f tfs[128];
    const int tid = threadIdx.x;
    if (tid < 128) {
        tfs[tid] = reinterpret_cast<const v4f*>(tf)[tid];
    }
    __syncthreads();

    const int p = blockIdx.x * 256 + tid;   // p = x*RH + y, grid sized exactly
    const int x = p / RH;
    const int y = p - x * RH;

    const float cx = cam[0], cy = cam[1], cz = cam[2];

    // ---- camera basis (matches reference: up0 = (0,1,0)); wave-uniform -> SALU
    float vdx = -cx, vdy = -cy, vdz = -cz;
    float inl = 1.0f / sqrtf(vdx * vdx + vdy * vdy + vdz * vdz);
    vdx *= inl; vdy *= inl; vdz *= inl;

    // right = normalize(cross(view_dir, up0)) = normalize((-vdz, 0, vdx))
    float rx = -vdz, rz = vdx;
    float irl = 1.0f / sqrtf(rx * rx + rz * rz);
    rx *= irl; rz *= irl;

    // up = normalize(cross(right, view_dir))   (right.y == 0)
    float ux = -rz * vdy;
    float uy = rz * vdx - rx * vdz;
    float uz = rx * vdy;
    float iul = 1.0f / sqrtf(ux * ux + uy * uy + uz * uz);
    ux *= iul; uy *= iul; uz *= iul;

    const float FOV_RAD = 0.5235987755982988f;  // radians(30)
    const float NEARP   = 0.1f;
    const float near_h  = 2.0f * tanf(FOV_RAD) * NEARP;
    const float near_w  = near_h * ((float)RW / (float)RH);

    const float uu = ((x + 0.5f) / (float)RW - 0.5f) * near_w;
    const float vv = ((y + 0.5f) / (float)RH - 0.5f) * near_h;

    // ray direction = normalize(NEAR*view + uu*right + vv*up)
    float rdx = NEARP * vdx + uu * rx + vv * ux;
    float rdy = NEARP * vdy           + vv * uy;
    float rdz = NEARP * vdz + uu * rz + vv * uz;
    float irdl = 1.0f / sqrtf(rdx * rdx + rdy * rdy + rdz * rdz);
    rdx *= irdl; rdy *= irdl; rdz *= irdl;

    // ---- slab test against [-1,1]^3
    const float ifx = 1.0f / rdx, ify = 1.0f / rdy, ifz = 1.0f / rdz;
    const float tx1 = (-1.0f - cx) * ifx, tx2 = (1.0f - cx) * ifx;
    const float ty1 = (-1.0f - cy) * ify, ty2 = (1.0f - cy) * ify;
    const float tz1 = (-1.0f - cz) * ifz, tz2 = (1.0f - cz) * ifz;
    const float tmin = fmaxf(fmaxf(fminf(tx1, tx2), fminf(ty1, ty2)), fminf(tz1, tz2));
    const float tmax = fminf(fminf(fmaxf(tx1, tx2), fmaxf(ty1, ty2)), fmaxf(tz1, tz2));
    const bool  hit  = (tmax >= 0.0f) && (tmin <= tmax);

    if (!hit) {
        v4f z4 = {0.0f, 0.0f, 0.0f, 0.0f};
        __builtin_nontemporal_store(z4, reinterpret_cast<v4f*>(out) + p);
        return;
    }

    const float tstart = fmaxf(tmin, 0.0f);
    const float dt     = fmaxf(tmax - tstart, 0.0f) * (1.0f / (float)NSAMP);
    const float scale  = 319.0f - 1e-4f;  // (res - 1 - 1e-4) in f32, as reference
    const float hs     = 0.5f * scale;

    // Fold the whole pos->voxel affine map per axis:
    //   clamp(0.5*(c + t*rd) + 0.5, 0, 1) * scale == clamp(t*(rd*hs) + (c*hs + hs), 0, scale)
    // One v_fma + one v_med3 per axis per sample.
    const float Ax = rdx * hs, Bx = fmaf(cx, hs, hs);
    const float Ay = rdy * hs, By = fmaf(cy, hs, hs);
    const float Az = rdz * hs, Bz = fmaf(cz, hs, hs);

    float r = 0.0f, g = 0.0f, b = 0.0f, alpha = 0.0f;

    // Samples are independent except for the cheap accumulate: unroll 8 keeps
    // ~32 b64 gathers in flight per wave (L2-latency hiding; LOADcnt is 6-bit).
#pragma unroll 8
    for (int k = 0; k < NSAMP; ++k) {
        const float t = fmaf((float)k + 0.5f, dt, tstart);

        const float txv = clampf(fmaf(t, Ax, Bx), 0.0f, scale);
        const float tyv = clampf(fmaf(t, Ay, By), 0.0f, scale);
        const float tzv = clampf(fmaf(t, Az, Bz), 0.0f, scale);

        const int xi = (int)txv;         // non-negative: truncation == floor
        const int yi = (int)tyv;
        const int zi = (int)tzv;
        const float fx = txv - (float)xi;
        const float fy = tyv - (float)yi;
        const float fz = tzv - (float)zi;

        // hi index is always lo+1 (coord <= 318.9999 < 319): 8 gathers off one
        // base; the z / z+1 pairs fuse into 4x global_load_b64.
        const float* bp = vol + ((xi * VRES + yi) * VRES + zi);
        const float v000 = bp[0];
        const float v001 = bp[1];
        const float v010 = bp[VRES];
        const float v011 = bp[VRES + 1];
        const float v100 = bp[VRES2];
        const float v101 = bp[VRES2 + 1];
        const float v110 = bp[VRES2 + VRES];
        const float v111 = bp[VRES2 + VRES + 1];

        const float c00 = lerpf(v000, v100, fx);
        const float c10 = lerpf(v010, v110, fx);
        const float c01 = lerpf(v001, v101, fx);
        const float c11 = lerpf(v011, v111, fx);
        const float c0  = lerpf(c00, c10, fy);
        const float c1  = lerpf(c01, c11, fy);
        const float intensity = lerpf(c0, c1, fz);

        // transfer-function lookup from LDS (2x ds_load_b128)
        const float tt = clampf(intensity * 127.0f, 0.0f, 127.0f);
        const int   li = (int)tt;
        const int   hi = min(li + 1, 127);
        const float fr = tt - (float)li;
        const v4f s0 = tfs[li];
        const v4f s1 = tfs[hi];
        const float cr = lerpf(s0.x, s1.x, fr);
        const float cg = lerpf(s0.y, s1.y, fr);
        const float cb = lerpf(s0.z, s1.z, fr);
        const float ca = lerpf(s0.w, s1.w, fr);

        // front-to-back compositing
        const float w = (1.0f - alpha) * ca;
        r = fmaf(w, cr, r);
        g = fmaf(w, cg, g);
        b = fmaf(w, cb, b);
        alpha += w;
    }

    // BG == 0, so rgb unchanged. [W,H,4] layout -> v4f index p = x*RH + y.
    // Non-temporal: framebuffer is write-once, keep it out of the L2 that
    // holds the (fully resident, 131MB) volume working set.
    v4f o = {r, g, b, alpha};
    __builtin_nontemporal_store(o, reinterpret_cast<v4f*>(out) + p);
}

extern "C" void kernel_launch(void* const* d_in, const int* in_sizes, int n_in,
                              void* d_out, int out_size, void* d_ws, size_t ws_size,
                              hipStream_t stream) {
    (void)in_sizes; (void)n_in; (void)d_ws; (void)ws_size; (void)out_size;
    const float* vol = (const float*)d_in[0];   // volume  [320,320,320] f32
    const float* tf  = (const float*)d_in[1];   // tf_tex  [128,4]       f32
    const float* cam = (const float*)d_in[2];   // cam_pos [3]           f32
    float* out = (float*)d_out;                 // [320,240,4] f32

    const int total  = RW * RH;                 // 76800 = 300 * 256 exactly
    const int block  = 256;                     // 8 wave32 waves per block
    const int grid   = total / block;
    Raycaster_22101901705954_kernel<<<grid, block, 0, stream>>>(vol, tf, cam, out);
}